// Attention_20126216749601
// MI455X (gfx1250) — compile-verified
//
#include <hip/hip_runtime.h>

typedef __attribute__((ext_vector_type(2))) float v2f;
typedef __attribute__((ext_vector_type(8))) float v8f;

#define DEC_HID 1024
#define ENC_HID 2048
#define SEQ_LEN 2048
#define BATCH   32

// ---------------- kernel 0: dec_dot[b] = dec[b,:]·w_dec + bias ----------------
__global__ void dec_dot_kernel(const float* __restrict__ dec,
                               const float* __restrict__ W,
                               const float* __restrict__ bias,
                               float* __restrict__ dec_dot) {
  __shared__ float red[256];
  int b = blockIdx.x, t = threadIdx.x;
  float acc = 0.f;
#pragma unroll
  for (int i = 0; i < DEC_HID; i += 256)
    acc += dec[b * DEC_HID + t + i] * W[t + i];
  red[t] = acc;
  __syncthreads();
  for (int off = 128; off > 0; off >>= 1) {
    if (t < off) red[t] += red[t + off];
    __syncthreads();
  }
  if (t == 0) dec_dot[b] = red[0] + bias[0];
}

// ---------------- kernel 1: score[b][s] = relu(enc_row·w_enc + dec_dot[b]) ----
// one wave (32 lanes) per flattened row r = s*BATCH + b; float4 coalesced loads
__global__ void score_kernel(const float* __restrict__ enc,
                             const float* __restrict__ w_enc,
                             const float* __restrict__ dec_dot,
                             float* __restrict__ score) {
  int wave = threadIdx.x >> 5;
  int lane = threadIdx.x & 31;
  int row  = blockIdx.x * 8 + wave;            // row = s*BATCH + b
  const float4* rowp = (const float4*)(enc + (size_t)row * ENC_HID);
  const float4* wp   = (const float4*)w_enc;
  float acc = 0.f;
#pragma unroll
  for (int i = 0; i < ENC_HID / 4; i += 32) {
    float4 a = rowp[i + lane];
    float4 w = wp[i + lane];
    acc += a.x * w.x + a.y * w.y + a.z * w.z + a.w * w.w;
  }
  for (int off = 16; off > 0; off >>= 1) acc += __shfl_down(acc, off);
  if (lane == 0) {
    int s = row >> 5;                          // row / BATCH
    int b = row & 31;                          // row % BATCH
    float v = acc + dec_dot[b];
    score[b * SEQ_LEN + s] = v > 0.f ? v : 0.f;
  }
}

// ---------------- kernel 2: softmax over s, per b ----------------------------
__global__ void softmax_kernel(const float* __restrict__ score,
                               float* __restrict__ attn) {
  __shared__ float red[256];
  int b = blockIdx.x, t = threadIdx.x;
  const float* sc = score + b * SEQ_LEN;
  float v[8];
  float m = -1e30f;
#pragma unroll
  for (int i = 0; i < 8; i++) { v[i] = sc[t + i * 256]; m = fmaxf(m, v[i]); }
  red[t] = m;
  __syncthreads();
  for (int off = 128; off > 0; off >>= 1) {
    if (t < off) red[t] = fmaxf(red[t], red[t + off]);
    __syncthreads();
  }
  m = red[0];
  __syncthreads();
  float e[8];
  float sum = 0.f;
#pragma unroll
  for (int i = 0; i < 8; i++) { e[i] = __expf(v[i] - m); sum += e[i]; }
  red[t] = sum;
  __syncthreads();
  for (int off = 128; off > 0; off >>= 1) {
    if (t < off) red[t] += red[t + off];
    __syncthreads();
  }
  float inv = 1.f / red[0];
#pragma unroll
  for (int i = 0; i < 8; i++) attn[b * SEQ_LEN + t + i * 256] = e[i] * inv;
}

// ---------------- kernel 3: context[b,e] = Σ_s attn[b,s]·enc[s,b,e] ----------
// WMMA f32 16x16x4: A[m,k] = attn[s0+k] broadcast over m (all D rows equal),
// B[k,n] = enc[s0+k, b, e0+n]. Lane>>4 selects the K-half (K = kh, kh+1 in the
// two A/B VGPR slots); identical slot mapping on A and B keeps K aligned.
//
// Each block owns a 32-float (128 B) e-slab so every HBM/L2 line it touches is
// fully consumed inside the block: waves 0-3 -> low 16-e tile, waves 4-7 ->
// high tile; each quartet splits the s range 4 ways (512 s each).
__global__ void context_kernel(const float* __restrict__ enc,
                               const float* __restrict__ attn,
                               float* __restrict__ out) {
  __shared__ float part[8 * 16];
  int b     = blockIdx.x >> 6;                 // 64 e-groups of 32
  int e0    = (blockIdx.x & 63) << 5;
  int wave  = threadIdx.x >> 5;
  int lane  = threadIdx.x & 31;
  int n     = lane & 15;
  int kh    = (lane >> 4) << 1;                // 0 or 2
  int tile  = wave >> 2;                       // 0: e0..e0+15, 1: e0+16..e0+31
  int sbase = (wave & 3) * (SEQ_LEN / 4);      // 512 s per wave

  const float* encp = enc + (size_t)b * ENC_HID + e0 + tile * 16 + n;
  const float* ap   = attn + b * SEQ_LEN;

  v8f acc = {};
#pragma unroll 4
  for (int s = sbase; s < sbase + SEQ_LEN / 4; s += 4) {
    v2f a = *(const v2f*)(ap + s + kh);        // attn[s+kh], attn[s+kh+1]
    v2f bm;
    bm.x = encp[(size_t)(s + kh + 0) * (BATCH * ENC_HID)];
    bm.y = encp[(size_t)(s + kh + 1) * (BATCH * ENC_HID)];
    acc = __builtin_amdgcn_wmma_f32_16x16x4_f32(
        /*neg_a=*/false, a, /*neg_b=*/false, bm,
        /*c_mod=*/(short)0, acc, /*reuse_a=*/false, /*reuse_b=*/false);
  }
  // D row 0 lives in lanes 0..15, VGPR 0 (all rows equal due to broadcast A)
  if (lane < 16) part[wave * 16 + n] = acc[0];
  __syncthreads();
  if (threadIdx.x < 32) {
    int t = threadIdx.x;                       // t<16: tile0, t>=16: tile1
    float sum = 0.f;
#pragma unroll
    for (int w = 0; w < 4; w++) sum += part[((t >> 4) * 4 + w) * 16 + (t & 15)];
    out[b * ENC_HID + e0 + t] = sum;
  }
}

// ---------------- launcher ---------------------------------------------------
extern "C" void kernel_launch(void* const* d_in, const int* in_sizes, int n_in,
                              void* d_out, int out_size, void* d_ws, size_t ws_size,
                              hipStream_t stream) {
  (void)in_sizes; (void)n_in; (void)out_size; (void)ws_size;
  const float* dec  = (const float*)d_in[0];   // (1,32,1024)
  const float* enc  = (const float*)d_in[1];   // (2048,32,2048)
  const float* W    = (const float*)d_in[2];   // (1,3072)
  const float* bias = (const float*)d_in[3];   // (1,)

  float* ws      = (float*)d_ws;
  float* dec_dot = ws;                          // 32 floats
  float* score   = ws + 32;                     // 32*2048 floats
  float* attn    = score + BATCH * SEQ_LEN;     // 32*2048 floats
  float* out     = (float*)d_out;               // (1,32,2048)

  dec_dot_kernel<<<BATCH, 256, 0, stream>>>(dec, W, bias, dec_dot);
  score_kernel<<<(SEQ_LEN * BATCH) / 8, 256, 0, stream>>>(enc, W + DEC_HID, dec_dot, score);
  softmax_kernel<<<BATCH, 256, 0, stream>>>(score, attn);
  context_kernel<<<BATCH * (ENC_HID / 32), 256, 0, stream>>>(enc, attn, out);
}